// GraphSage_44298292691347
// MI455X (gfx1250) — compile-verified
//
#include <hip/hip_runtime.h>

#define DF 64

typedef __attribute__((ext_vector_type(2))) float v2f;
typedef __attribute__((ext_vector_type(8))) float v8f;

static __device__ __forceinline__ v8f wmma_f32(v2f a, v2f b, v8f c) {
  // D(16x16,f32) = A(16x4,f32) * B(4x16,f32) + C
  return __builtin_amdgcn_wmma_f32_16x16x4_f32(
      /*neg_a=*/false, a, /*neg_b=*/false, b,
      /*c_mod=*/(short)0, c, /*reuse_a=*/false, /*reuse_b=*/false);
}

// -------------------------------------------------------------------------
// Zero-fill (agg buffer must start at 0.0f == bit pattern 0 for umax trick)
// -------------------------------------------------------------------------
__global__ void zero_f32(float* __restrict__ p, int n) {
  int i = blockIdx.x * blockDim.x + threadIdx.x;
  if (i < n) p[i] = 0.0f;
}

// -------------------------------------------------------------------------
// Fused edge kernel: m = relu(h[src] @ W_pool + b_pool) ; agg[dst] = max(...)
// One wave handles a 16-edge x 64-col tile with 64 v_wmma_f32_16x16x4_f32.
// Messages are >= 0 after relu, agg init 0 -> atomic umax on float bits.
// -------------------------------------------------------------------------
__global__ __launch_bounds__(256) void edge_pool_max(
    const float* __restrict__ h, const int* __restrict__ src,
    const int* __restrict__ dst, const float* __restrict__ Wp,
    const float* __restrict__ bp, unsigned int* __restrict__ agg,
    int n_edges) {
  __shared__ float w_lds[DF * DF];
  __shared__ float b_lds[DF];
  for (int i = threadIdx.x; i < DF * DF; i += 256) w_lds[i] = Wp[i];
  if (threadIdx.x < DF) b_lds[threadIdx.x] = bp[threadIdx.x];
  __syncthreads();

  const int lane = threadIdx.x & 31;
  const int wave = threadIdx.x >> 5;
  const int half = lane >> 4;   // 0: K sub {0,1}, rows M 0..7 ; 1: K sub {2,3}, rows M 8..15
  const int l16  = lane & 15;
  const int coff = half * 2;

  // Hoist loop-invariant B fragments (W_pool) into registers: 16 K-chunks x 4 N-tiles.
  v2f bf[16][4];
#pragma unroll
  for (int k = 0; k < 16; ++k) {
#pragma unroll
    for (int t = 0; t < 4; ++t) {
      const int col = t * 16 + l16;
      v2f b;
      b.x = w_lds[(4 * k + coff + 0) * DF + col];
      b.y = w_lds[(4 * k + coff + 1) * DF + col];
      bf[k][t] = b;
    }
  }
  float bcol[4];
#pragma unroll
  for (int t = 0; t < 4; ++t) bcol[t] = b_lds[t * 16 + l16];

  const int n_tiles = (n_edges + 15) >> 4;
  for (int tile = blockIdx.x * 8 + wave; tile < n_tiles; tile += gridDim.x * 8) {
    const int e0 = tile * 16;
    const int eA = e0 + l16;
    const int eC = (eA < n_edges) ? eA : 0;     // clamp (avoids divergence at WMMA)
    const int sA = src[eC];
    const int dA = dst[eC];
    const float* rp = h + (size_t)sA * DF + coff;

    v8f c[4] = {v8f{}, v8f{}, v8f{}, v8f{}};
#pragma unroll
    for (int k = 0; k < 16; ++k) {
      v2f a = *(const v2f*)(rp + 4 * k);        // 8B-aligned global_load_b64
      c[0] = wmma_f32(a, bf[k][0], c[0]);
      c[1] = wmma_f32(a, bf[k][1], c[1]);
      c[2] = wmma_f32(a, bf[k][2], c[2]);
      c[3] = wmma_f32(a, bf[k][3], c[3]);
    }

    // bias + relu + atomic max scatter. C layout: VGPR r = row (r + 8*half), col = t*16+l16.
#pragma unroll
    for (int t = 0; t < 4; ++t) {
      const int col = t * 16 + l16;
#pragma unroll
      for (int r = 0; r < 8; ++r) {
        const int M = r + half * 8;
        float v = c[t][r] + bcol[t];
        v = v > 0.0f ? v : 0.0f;
        const int e = e0 + M;
        const int d = __shfl(dA, M, 32);        // dst of row M lives in lane M
        if (e < n_edges && v > 0.0f) {
          atomicMax(agg + (size_t)d * DF + col, __float_as_uint(v));
        }
      }
    }
  }
}

// -------------------------------------------------------------------------
// Node kernel: out = h @ W_self + agg @ W_neigh + bias ; optional relu.
// One wave: 16-node tile, 128 WMMAs (2 GEMMs x 4 N-tiles x 16 K-chunks).
// -------------------------------------------------------------------------
__global__ __launch_bounds__(256) void node_update(
    const float* __restrict__ h, const float* __restrict__ agg,
    const float* __restrict__ Ws, const float* __restrict__ Wn,
    const float* __restrict__ bias, float* __restrict__ out,
    int n_nodes, int do_relu) {
  __shared__ float ws_lds[DF * DF];
  __shared__ float wn_lds[DF * DF];
  __shared__ float b_lds[DF];
  for (int i = threadIdx.x; i < DF * DF; i += 256) {
    ws_lds[i] = Ws[i];
    wn_lds[i] = Wn[i];
  }
  if (threadIdx.x < DF) b_lds[threadIdx.x] = bias[threadIdx.x];
  __syncthreads();

  const int lane = threadIdx.x & 31;
  const int wave = threadIdx.x >> 5;
  const int half = lane >> 4;
  const int l16  = lane & 15;
  const int coff = half * 2;

  float bcol[4];
#pragma unroll
  for (int t = 0; t < 4; ++t) bcol[t] = b_lds[t * 16 + l16];

  const int n_tiles = (n_nodes + 15) >> 4;
  for (int tile = blockIdx.x * 8 + wave; tile < n_tiles; tile += gridDim.x * 8) {
    const int e0 = tile * 16;
    const int row = e0 + l16;
    const int rowc = (row < n_nodes) ? row : 0;
    const float* rp1 = h + (size_t)rowc * DF + coff;
    const float* rp2 = agg + (size_t)rowc * DF + coff;

    v8f c[4] = {v8f{}, v8f{}, v8f{}, v8f{}};
#pragma unroll
    for (int k = 0; k < 16; ++k) {
      v2f a1 = *(const v2f*)(rp1 + 4 * k);
      v2f a2 = *(const v2f*)(rp2 + 4 * k);
#pragma unroll
      for (int t = 0; t < 4; ++t) {
        const int col = t * 16 + l16;
        v2f b1, b2;
        b1.x = ws_lds[(4 * k + coff + 0) * DF + col];
        b1.y = ws_lds[(4 * k + coff + 1) * DF + col];
        b2.x = wn_lds[(4 * k + coff + 0) * DF + col];
        b2.y = wn_lds[(4 * k + coff + 1) * DF + col];
        c[t] = wmma_f32(a1, b1, c[t]);
        c[t] = wmma_f32(a2, b2, c[t]);
      }
    }

#pragma unroll
    for (int t = 0; t < 4; ++t) {
      const int col = t * 16 + l16;
#pragma unroll
      for (int r = 0; r < 8; ++r) {
        const int M = r + half * 8;
        const int nrow = e0 + M;
        float v = c[t][r] + bcol[t];
        if (do_relu) v = v > 0.0f ? v : 0.0f;
        if (nrow < n_nodes) out[(size_t)nrow * DF + col] = v;
      }
    }
  }
}

// -------------------------------------------------------------------------
// Launcher: 3 layers x (zero agg, fused edge-MLP+max, node update).
// Workspace: [agg 12.8MB][h1 12.8MB][h2 12.8MB].
// -------------------------------------------------------------------------
extern "C" void kernel_launch(void* const* d_in, const int* in_sizes, int n_in,
                              void* d_out, int out_size, void* d_ws, size_t ws_size,
                              hipStream_t stream) {
  const float* in_feat = (const float*)d_in[0];
  const int*   src     = (const int*)d_in[1];
  const int*   dst     = (const int*)d_in[2];
  const float* W_pool  = (const float*)d_in[3];
  const float* b_pool  = (const float*)d_in[4];
  const float* W_self  = (const float*)d_in[5];
  const float* W_neigh = (const float*)d_in[6];
  const float* bias    = (const float*)d_in[7];
  float* out = (float*)d_out;

  const int n_nodes = in_sizes[0] / DF;   // 50000
  const int n_edges = in_sizes[1];        // 1250000
  const size_t feat_bytes = (size_t)n_nodes * DF * sizeof(float);

  char* ws = (char*)d_ws;
  float*        aggf = (float*)ws;
  unsigned int* aggu = (unsigned int*)ws;
  float* h1 = (float*)(ws + feat_bytes);
  float* h2 = (float*)(ws + 2 * feat_bytes);

  const int nz = n_nodes * DF;
  const int etiles = (n_edges + 15) / 16;
  int eblocks = (etiles + 7) / 8; if (eblocks > 4096) eblocks = 4096;
  const int ntiles = (n_nodes + 15) / 16;
  int nblocks = (ntiles + 7) / 8; if (nblocks > 4096) nblocks = 4096;

  const float* hcur = in_feat;
  for (int l = 0; l < 3; ++l) {
    zero_f32<<<(nz + 255) / 256, 256, 0, stream>>>(aggf, nz);

    edge_pool_max<<<eblocks, 256, 0, stream>>>(
        hcur, src, dst, W_pool + (size_t)l * DF * DF, b_pool + (size_t)l * DF,
        aggu, n_edges);

    float* hnext = (l == 2) ? out : ((l == 0) ? h1 : h2);
    node_update<<<nblocks, 256, 0, stream>>>(
        hcur, aggf, W_self + (size_t)l * DF * DF, W_neigh + (size_t)l * DF * DF,
        bias + (size_t)l * DF, hnext, n_nodes, (l < 2) ? 1 : 0);
    hcur = hnext;
  }
}